// MonarchFactor_60052232732950
// MI455X (gfx1250) — compile-verified
//
#include <hip/hip_runtime.h>

typedef __attribute__((ext_vector_type(2))) float v2f;
typedef __attribute__((ext_vector_type(4))) float v4f;
typedef __attribute__((ext_vector_type(8))) float v8f;

#define NBLOCKS   4
#define BLK_R     16
#define IN_BLK    1024
#define IN_FEAT   4096
#define OUT_FEAT  64
#define WAVES_PER_WG 8
#define THREADS   (WAVES_PER_WG * 32)

// One wave owns one 16-token tile. For each of the 4 diagonal blocks:
//  - workgroup stages the 16x1024 fp32 weight block into LDS (64 KB)
//  - wave runs K=1024 as 256 chained v_wmma_f32_16x16x4_f32 accumulations
//  - A operand: float2 per lane straight from global x (layout matches ISA A map)
//  - B operand: float2 per lane from LDS (layout matches ISA B map)
__global__ __launch_bounds__(THREADS)
void monarch_wmma_f32_kernel(const float* __restrict__ x,
                             const float* __restrict__ w,
                             float* __restrict__ out,
                             int ntiles) {
    __shared__ float lds_w[BLK_R * IN_BLK];   // 64 KB weight block

    const int tid  = threadIdx.x;
    const int wave = tid >> 5;
    const int lane = tid & 31;
    const int n    = lane & 15;   // A: token row within tile | B/D: column r
    const int kh   = lane >> 4;   // K half-select (0: K=0,1  1: K=2,3)

    const int tile = blockIdx.x * WAVES_PER_WG + wave;

    for (int blk = 0; blk < NBLOCKS; ++blk) {
        // ---- stage weight block into LDS (16384 floats, 16 x float4 per thread)
        {
            const v4f* __restrict__ wsrc = (const v4f*)(w + (size_t)blk * BLK_R * IN_BLK);
            v4f* wdst = (v4f*)lds_w;
            #pragma unroll
            for (int i = 0; i < (BLK_R * IN_BLK / 4) / THREADS; ++i) {
                wdst[tid + i * THREADS] = wsrc[tid + i * THREADS];
            }
        }
        __syncthreads();

        if (tile < ntiles) {
            const float* __restrict__ xrow =
                x + (size_t)(tile * 16 + n) * IN_FEAT + blk * IN_BLK + 2 * kh;
            const float* __restrict__ wrow = lds_w + n * IN_BLK + 2 * kh;

            v8f acc = {};
            #pragma unroll 8
            for (int k = 0; k < IN_BLK; k += 4) {
                v2f a = *(const v2f*)(xrow + k);
                v2f b = *(const v2f*)(wrow + k);
                // 8 args: (neg_a, A, neg_b, B, c_mod, C, reuse_a, reuse_b)
                acc = __builtin_amdgcn_wmma_f32_16x16x4_f32(
                        false, a, false, b, (short)0, acc, false, false);
            }

            // D layout: vgpr j -> token (j + 8*kh), column n
            float* __restrict__ obase =
                out + (size_t)tile * 16 * OUT_FEAT + (size_t)(8 * kh) * OUT_FEAT
                    + blk * BLK_R + n;
            #pragma unroll
            for (int j = 0; j < 8; ++j) {
                obase[j * OUT_FEAT] = acc[j];
            }
        }
        __syncthreads();   // protect LDS before next block's staging
    }
}

extern "C" void kernel_launch(void* const* d_in, const int* in_sizes, int n_in,
                              void* d_out, int out_size, void* d_ws, size_t ws_size,
                              hipStream_t stream) {
    const float* x = (const float*)d_in[0];
    const float* w = (const float*)d_in[1];
    float* out = (float*)d_out;

    const int tokens = in_sizes[0] / IN_FEAT;   // B*S = 32768
    const int ntiles = tokens / 16;             // 2048
    const int nblocksx = (ntiles + WAVES_PER_WG - 1) / WAVES_PER_WG;  // 256

    monarch_wmma_f32_kernel<<<nblocksx, THREADS, 0, stream>>>(x, w, out, ntiles);
}